// MMCL_75874892251533
// MI455X (gfx1250) — compile-verified
//
#include <hip/hip_runtime.h>
#include <math.h>
#include <stdint.h>

// ---------------------------------------------------------------------------
// Hard-negative contrastive loss, MI455X (gfx1250).
//   B=512 rows, N=65536 logits/row, K=655 hard negatives, TEMP=10.
// One workgroup per row. The whole 256KB row is staged HBM -> LDS once via
// CDNA5 async loads (global_load_async_to_lds_b128 + s_wait_asynccnt), then
// a 3-level LDS radix-select finds the exact K-th largest non-target value,
// and a final LDS sweep accumulates the stable softmax denominator.
// ---------------------------------------------------------------------------

namespace {
constexpr int   NLOG = 65536;   // logits per row
constexpr int   TPB  = 512;     // 16 wave32s per workgroup
constexpr int   KSEL = 655;     // hard negatives per row
constexpr float TEMP = 10.0f;

// LDS layout (dynamic shared, base offset 0 since there is no static LDS):
//   [0,            262144)  row data, 65536 f32
//   [262144,       270336)  histogram, 2048 u32
//   [270336,       272384)  chunk prefix sums, 512 u32
//   [272384,       272416)  broadcast slots, 8 u32
//   [272416,       274464)  float reduction scratch, 512 f32
constexpr size_t OFF_HIST   = (size_t)NLOG * 4;
constexpr size_t OFF_CSUM   = OFF_HIST + 2048u * 4u;
constexpr size_t OFF_UBC    = OFF_CSUM + (size_t)TPB * 4u;
constexpr size_t OFF_FRED   = OFF_UBC + 8u * 4u;
constexpr size_t SMEM_BYTES = OFF_FRED + (size_t)TPB * 4u;   // 274,464 B < 320 KB
} // namespace

// Monotonic float -> u32 key (larger float <=> larger key).
__device__ __forceinline__ unsigned fkey(float f) {
  unsigned u = __float_as_uint(f);
  return (u & 0x80000000u) ? ~u : (u | 0x80000000u);
}
__device__ __forceinline__ float kinv(unsigned u) {
  return __uint_as_float((u & 0x80000000u) ? (u ^ 0x80000000u) : ~u);
}

// Find, scanning bins from the TOP (descending), the bin b* such that
//   countAbove(b*) < need <= countAbove(b*) + hist[b*].
// Parallel: per-thread chunk sums + Hillis-Steele inclusive scan, then the
// unique winning thread resolves within its chunk. Returns bin and
// count-strictly-above-bin. All TPB threads must call this uniformly.
__device__ void radix_select_bin(const unsigned* hist, int NB, unsigned need,
                                 unsigned* csum, unsigned* ubc,
                                 unsigned& outBin, unsigned& outAbove) {
  const int t = threadIdx.x;
  const int C = NB / TPB;              // bins per chunk (4 or 2)
  const int hiBin = NB - t * C;        // chunk t covers [hiBin-C, hiBin), chunk 0 = top
  unsigned s = 0u;
  for (int j = hiBin - C; j < hiBin; ++j) s += hist[j];
  csum[t] = s;
  __syncthreads();
  for (int off = 1; off < TPB; off <<= 1) {
    unsigned add = (t >= off) ? csum[t - off] : 0u;
    __syncthreads();
    csum[t] += add;
    __syncthreads();
  }
  const unsigned above = (t == 0) ? 0u : csum[t - 1];
  const unsigned inc   = csum[t];
  if (above < need && need <= inc) {   // unique winner (need >= 1, csum monotone)
    unsigned a = above;
    for (int j = hiBin - 1; j >= hiBin - C; --j) {
      unsigned h = hist[j];
      if (a + h >= need) { ubc[0] = (unsigned)j; ubc[1] = a; break; }
      a += h;
    }
  }
  __syncthreads();
  outBin = ubc[0];
  outAbove = ubc[1];
  __syncthreads();
}

__global__ __launch_bounds__(TPB, 1)
void hn_rows_kernel(const float* __restrict__ logits,
                    const int* __restrict__ targets,
                    float* __restrict__ row_loss) {
  extern __shared__ unsigned char smem[];
  float*    df   = reinterpret_cast<float*>(smem);
  float4*   df4  = reinterpret_cast<float4*>(smem);
  unsigned* hist = reinterpret_cast<unsigned*>(smem + OFF_HIST);
  unsigned* csum = reinterpret_cast<unsigned*>(smem + OFF_CSUM);
  unsigned* ubc  = reinterpret_cast<unsigned*>(smem + OFF_UBC);
  float*    fred = reinterpret_cast<float*>(smem + OFF_FRED);

  const int t   = threadIdx.x;
  const int row = blockIdx.x;
  const int tgt = targets[row];
  const unsigned long long base =
      (unsigned long long)(logits + (size_t)row * NLOG);

  // ---- Stage the entire row HBM -> LDS with CDNA5 async copies. -----------
  // 16 B per lane per instruction; 32 instructions per thread cover 256 KB.
  // LDS dest offset == global byte offset because the data region sits at
  // dynamic-LDS base 0 (no static __shared__ in this kernel).
  constexpr int CPY_ITERS = (NLOG * 4) / (TPB * 16);   // 32
  for (int k = 0; k < CPY_ITERS; ++k) {
    unsigned off = (unsigned)((k * TPB + t) * 16);
    asm volatile("global_load_async_to_lds_b128 %0, %1, %2"
                 :: "v"(off), "v"(off), "s"(base) : "memory");
  }
  // Zero the level-1 histogram while the async copies are in flight.
  for (int i = t; i < 2048; i += TPB) hist[i] = 0u;
  asm volatile("s_wait_asynccnt 0" ::: "memory");
  __syncthreads();

  // ---- Pass 1: row max + level-1 histogram (key bits 31..21). -------------
  float lmax = -3.402823466e38f;
  for (int q = t; q < NLOG / 4; q += TPB) {
    float4 v = df4[q];
    const int e = q * 4;
    float vv[4] = {v.x, v.y, v.z, v.w};
#pragma unroll
    for (int c = 0; c < 4; ++c) {
      lmax = fmaxf(lmax, vv[c]);
      if (e + c != tgt) atomicAdd(&hist[fkey(vv[c]) >> 21], 1u);
    }
  }
  fred[t] = lmax;
  __syncthreads();
  for (int s = TPB / 2; s > 0; s >>= 1) {
    if (t < s) fred[t] = fmaxf(fred[t], fred[t + s]);
    __syncthreads();
  }
  const float m = fred[0];   // row max; always a member of the candidate set
  __syncthreads();

  unsigned b1, a1;
  radix_select_bin(hist, 2048, (unsigned)KSEL, csum, ubc, b1, a1);
  const unsigned need2 = (unsigned)KSEL - a1;

  // ---- Pass 2: level-2 histogram (key bits 20..10) inside bin b1. ---------
  for (int i = t; i < 2048; i += TPB) hist[i] = 0u;
  __syncthreads();
  for (int q = t; q < NLOG / 4; q += TPB) {
    float4 v = df4[q];
    const int e = q * 4;
    float vv[4] = {v.x, v.y, v.z, v.w};
#pragma unroll
    for (int c = 0; c < 4; ++c) {
      if (e + c == tgt) continue;
      unsigned u = fkey(vv[c]);
      if ((u >> 21) == b1) atomicAdd(&hist[(u >> 10) & 0x7FFu], 1u);
    }
  }
  __syncthreads();

  unsigned b2, a2;
  radix_select_bin(hist, 2048, need2, csum, ubc, b2, a2);
  const unsigned need3 = need2 - a2;
  const unsigned top22 = (b1 << 11) | b2;

  // ---- Pass 3: level-3 histogram (key bits 9..0) inside top-22 match. -----
  for (int i = t; i < 1024; i += TPB) hist[i] = 0u;
  __syncthreads();
  for (int q = t; q < NLOG / 4; q += TPB) {
    float4 v = df4[q];
    const int e = q * 4;
    float vv[4] = {v.x, v.y, v.z, v.w};
#pragma unroll
    for (int c = 0; c < 4; ++c) {
      if (e + c == tgt) continue;
      unsigned u = fkey(vv[c]);
      if ((u >> 10) == top22) atomicAdd(&hist[u & 0x3FFu], 1u);
    }
  }
  __syncthreads();

  unsigned b3, a3;
  radix_select_bin(hist, 1024, need3, csum, ubc, b3, a3);
  const unsigned needEq = need3 - a3;         // copies of the threshold value
  const unsigned uthr   = (top22 << 10) | b3; // exact K-th largest key
  const float    vthr   = kinv(uthr);

  // ---- Pass 4: stable softmax denominator over the selected set. ----------
  float lsum = 0.0f;
  for (int q = t; q < NLOG / 4; q += TPB) {
    float4 v = df4[q];
    const int e = q * 4;
    float vv[4] = {v.x, v.y, v.z, v.w};
#pragma unroll
    for (int c = 0; c < 4; ++c) {
      if (e + c == tgt) continue;
      if (fkey(vv[c]) > uthr) lsum += expf(TEMP * (vv[c] - m));
    }
  }
  fred[t] = lsum;
  __syncthreads();
  for (int s = TPB / 2; s > 0; s >>= 1) {
    if (t < s) fred[t] += fred[t + s];
    __syncthreads();
  }
  if (t == 0) {
    const float tval = df[tgt];
    const float S = fred[0]
                  + (float)needEq * expf(TEMP * (vthr - m))
                  + expf(TEMP * (tval - m));
    row_loss[row] = logf(S) - TEMP * (tval - m);   // = -log_softmax[..., 0]
  }
}

__global__ __launch_bounds__(TPB, 1)
void hn_reduce_kernel(const float* __restrict__ rl, float* __restrict__ out,
                      int B) {
  __shared__ float sh[TPB];
  const int t = threadIdx.x;
  float s = 0.0f;
  for (int i = t; i < B; i += TPB) s += rl[i];
  sh[t] = s;
  __syncthreads();
  for (int k = TPB / 2; k > 0; k >>= 1) {
    if (t < k) sh[t] += sh[t + k];
    __syncthreads();
  }
  if (t == 0) out[0] = sh[0] / (float)B;
}

extern "C" void kernel_launch(void* const* d_in, const int* in_sizes, int n_in,
                              void* d_out, int out_size, void* d_ws,
                              size_t ws_size, hipStream_t stream) {
  const float* logits  = (const float*)d_in[0];
  const int*   targets = (const int*)d_in[1];
  const int B = in_sizes[1];                 // 512 rows
  float* row_loss = (float*)d_ws;            // B floats of scratch

  (void)hipFuncSetAttribute((const void*)hn_rows_kernel,
                            hipFuncAttributeMaxDynamicSharedMemorySize,
                            (int)SMEM_BYTES);

  hn_rows_kernel<<<dim3(B), dim3(TPB), SMEM_BYTES, stream>>>(logits, targets,
                                                             row_loss);
  hn_reduce_kernel<<<dim3(1), dim3(TPB), 0, stream>>>(row_loss, (float*)d_out,
                                                      B);
}